// ButterworthFilter_76003741270711
// MI455X (gfx1250) — compile-verified
//
#include <hip/hip_runtime.h>
#include <hip/hip_bf16.h>

// ---------------------------------------------------------------------------
// Butterworth filtfilt, 2048 rows x 16384 samples, f32, on gfx1250.
// Parallel-in-time exact chunk-scan IIR (DF2T, order-8 state):
//   z' = M z + c x is linear => per 514-sample chunk: z_out = A z_in + p,
//   A = M^514.
//   Phase1: per-lane chunk response p from zero state (scalar DF2T).
//   Phase2: log-depth Hillis-Steele affine scan across the 32 lanes; each
//           level's batched (8x8)x(8x32) matvec runs on
//           v_wmma_f32_16x16x4_f32 (2 N-tiles x 2 K-chunks = 4 WMMA/level),
//           with A^(2^l) pre-swizzled into the A-operand register layout.
//   Phase3: replay each chunk with its true entry state, emit outputs.
// Forward pass writes y into d_out (mid) + 54-float/row pad buffer in d_ws.
// Backward pass reads y reversed, overwrites d_out in place (slot-exclusive,
// read-before-write per lane), producing the final filtfilt result.
// ---------------------------------------------------------------------------

#define NROWS 2048
#define T0    16384
#define PPAD  27                      // padlen = 3 * ntaps, ntaps = 9
#define TEXT  (T0 + 2*PPAD)           // 16438
#define NCH   32                      // one chunk per lane (wave32)
#define CHL   ((TEXT + NCH - 1)/NCH)  // 514 (32*514 = 16448 >= 16438)
#define PADW  (2*PPAD)                // 54
#define NS    8                       // filter state size (order 8)
#define NLVL  5                       // log2(NCH) scan levels
#define ASWZ_FLOATS (NLVL * 4 * 32)   // 5 levels x 4 A-operand regs x 32 lanes

typedef __attribute__((ext_vector_type(8))) float v8f;
typedef __attribute__((ext_vector_type(2))) float v2f;

// ---------------------------------------------------------------------------
// Setup kernel: R_l = (M^CHL)^(2^l), l = 0..4, in double via binary powering
// on 64 threads (one 8x8 element per thread, products via LDS), emitted
// pre-swizzled into the WMMA f32 16x16x4 A-operand layout:
//   lane m (m<16) holds (M=m, K=c*4+{0,1}); lane 16+m holds (M=m, K=c*4+{2,3})
// with rows M>=8 zero-padded.  M[i][j] = [j==i+1] - a[i+1]*[j==0].
// ---------------------------------------------------------------------------
__global__ void butter_setup(const float* __restrict__ a, float* __restrict__ Aswz) {
  __shared__ double Base[64], R[64];
  const int tid = threadIdx.x;          // blockDim.x == 64
  const int i = tid >> 3, j = tid & 7;

  double m = ((j == i + 1) ? 1.0 : 0.0) - (double)a[i + 1] * ((j == 0) ? 1.0 : 0.0);
  Base[tid] = m;
  R[tid]    = (i == j) ? 1.0 : 0.0;
  __syncthreads();

  int e = CHL;                          // R = M^CHL by binary powering
  while (e) {
    if (e & 1) {
      double acc = 0.0;
      #pragma unroll
      for (int k = 0; k < 8; ++k) acc += R[i * 8 + k] * Base[k * 8 + j];
      __syncthreads();
      R[tid] = acc;
      __syncthreads();
    }
    double acc2 = 0.0;
    #pragma unroll
    for (int k = 0; k < 8; ++k) acc2 += Base[i * 8 + k] * Base[k * 8 + j];
    __syncthreads();
    Base[tid] = acc2;
    __syncthreads();
    e >>= 1;
  }

  for (int l = 0; l < NLVL; ++l) {
    if (l > 0) {                        // R = R*R (next power of two)
      double acc = 0.0;
      #pragma unroll
      for (int k = 0; k < 8; ++k) acc += R[i * 8 + k] * R[k * 8 + j];
      __syncthreads();
      R[tid] = acc;
      __syncthreads();
    }
    // Emit 128 swizzled A-operand values for this level (2 per thread):
    // entry e2 = c*64 + r*32 + lane  ->  Aswz[(l*4 + c*2 + r)*32 + lane]
    #pragma unroll
    for (int half = 0; half < 2; ++half) {
      const int e2  = tid + half * 64;
      const int c   = e2 >> 6;          // K chunk (0: K=0..3, 1: K=4..7)
      const int r   = (e2 >> 5) & 1;    // reg within chunk
      const int lam = e2 & 31;          // lane
      const int M   = lam & 15;
      const int K   = c * 4 + r + ((lam >= 16) ? 2 : 0);
      const float v = (M < 8) ? (float)R[M * 8 + K] : 0.0f;
      Aswz[(l * 4 + c * 2 + r) * 32 + lam] = v;
    }
    __syncthreads();
  }
}

// ---------------------------------------------------------------------------
// DF2T step, matching the reference arithmetic:
//   y = b0*x + z0 ;  z_i = z_{i+1} + b_{i+1}*x - a_{i+1}*y  (z8 == 0)
// ---------------------------------------------------------------------------
__device__ __forceinline__ float df2t_step(float z[NS], const float br[NS],
                                           const float ar[NS], float b0, float xv) {
  float y = fmaf(b0, xv, z[0]);
  #pragma unroll
  for (int i = 0; i < NS - 1; ++i) {
    float t = fmaf(br[i], xv, z[i + 1]);
    z[i] = fmaf(-ar[i], y, t);
  }
  z[NS - 1] = fmaf(-ar[NS - 1], y, br[NS - 1] * xv);
  return y;
}

// ---------------------------------------------------------------------------
// One filtering pass over the extended sequence. MODE 0 = forward over
// reflect-padded x, writing y (mid -> ymid, pads -> ypad). MODE 1 = backward
// over reversed y, writing the final trimmed output into ymid in place.
// One wave per row; lane k owns samples [k*CHL, (k+1)*CHL).
// ---------------------------------------------------------------------------
template <int MODE>
__global__ void __launch_bounds__(256)
filt_pass(const float* __restrict__ x, const float* __restrict__ bvec,
          const float* __restrict__ avec, const float* __restrict__ zivec,
          float* __restrict__ ymid, float* __restrict__ ypad,
          const float* __restrict__ Aswz) {
  const int wave = threadIdx.x >> 5;
  const int lane = threadIdx.x & 31;
  const int row  = blockIdx.x * (blockDim.x >> 5) + wave;
  if (row >= NROWS) return;             // uniform per wave (grid covers exactly)

  const float* __restrict__ xrow = x    + (size_t)row * T0;
  float* __restrict__ yrow       = ymid + (size_t)row * T0;
  float* __restrict__ prow       = ypad + (size_t)row * PADW;

  const float b0 = bvec[0];
  float br[NS], ar[NS];
  #pragma unroll
  for (int i = 0; i < NS; ++i) { br[i] = bvec[i + 1]; ar[i] = avec[i + 1]; }

  // Source sample u of the (possibly reversed) extended sequence.
  auto load_s = [&](int u) -> float {
    if (MODE == 0) {
      if (u < PPAD) return 2.0f * xrow[0] - xrow[PPAD - u];
      int i = u - PPAD;
      if (i < T0) return xrow[i];
      return 2.0f * xrow[T0 - 1] - xrow[2 * (T0 - 1) - i];
    } else {
      int tt = TEXT - 1 - u;            // reversed read of y
      if (tt < PPAD) return prow[tt];
      if (tt < PPAD + T0) return yrow[tt - PPAD];
      return prow[PPAD + (tt - (PPAD + T0))];
    }
  };

  const int start = lane * CHL;

  // Warm the cache for this lane's stream (global_prefetch path).
  if (MODE == 0) {
    int i0 = start - PPAD; if (i0 < 0) i0 = 0; if (i0 >= T0) i0 = T0 - 1;
    __builtin_prefetch(xrow + i0, 0, 1);
  } else {
    int tt = TEXT - 1 - start - PPAD; if (tt < 0) tt = 0; if (tt >= T0) tt = T0 - 1;
    __builtin_prefetch(yrow + tt, 0, 1);
  }

  const float s0 = load_s(0);

  // ---- Phase 1: chunk response from zero state (uniform CHL, zero-padded
  //      tail for lane 31 whose p is never consumed). ----
  float p[NS];
  #pragma unroll
  for (int i = 0; i < NS; ++i) p[i] = 0.0f;
  for (int t = 0; t < CHL; ++t) {
    const int u = start + t;
    const float xv = (u < TEXT) ? load_s(u) : 0.0f;
    (void)df2t_step(p, br, ar, b0, xv);
  }

  // ---- Phase 2: log-depth affine scan. q_0 = zi*s0, q_k = p_{k-1};
  //      level l: s_k += A^(2^l) * s_{k-2^l}.  After NLVL levels lane k
  //      holds z_start(k) = A^k q_0 + sum_{j<k} A^(k-1-j) p_j exactly.
  //      The 32 batched 8x8 matvecs per level run as one padded
  //      (16x8)x(8x32) product on v_wmma_f32_16x16x4_f32. ----
  float s[NS];
  #pragma unroll
  for (int i = 0; i < NS; ++i) {
    float up = __shfl_up(p[i], 1, 32);
    s[i] = (lane == 0) ? (zivec[i] * s0) : up;
  }
  const bool lo = (lane < 16);
  #pragma unroll
  for (int l = 0; l < NLVL; ++l) {
    const int o = 1 << l;
    float t[NS], tsw[NS];
    #pragma unroll
    for (int i = 0; i < NS; ++i) t[i] = __shfl_up(s[i], o, 32);
    #pragma unroll
    for (int i = 0; i < NS; ++i) tsw[i] = __shfl_xor(t[i], 16, 32);

    // A operand: pre-swizzled A^(2^l) (chunk c regs {0,1} per lane).
    const float* at = Aswz + l * 4 * 32;
    v2f A0, A1;
    A0[0] = at[0 * 32 + lane];  A0[1] = at[1 * 32 + lane];
    A1[0] = at[2 * 32 + lane];  A1[1] = at[3 * 32 + lane];

    // B operand: tile0 = columns(lanes) 0..15, tile1 = columns 16..31.
    // Layout: lanes 0-15 hold K=c*4+{0,1}, lanes 16-31 hold K=c*4+{2,3}.
    v2f B00, B01, B10, B11;
    B00[0] = lo ? t[0]   : tsw[2];  B00[1] = lo ? t[1]   : tsw[3];
    B01[0] = lo ? t[4]   : tsw[6];  B01[1] = lo ? t[5]   : tsw[7];
    B10[0] = lo ? tsw[0] : t[2];    B10[1] = lo ? tsw[1] : t[3];
    B11[0] = lo ? tsw[4] : t[6];    B11[1] = lo ? tsw[5] : t[7];

    v8f d0 = {}, d1 = {};
    d0 = __builtin_amdgcn_wmma_f32_16x16x4_f32(false, A0, false, B00, (short)0, d0, false, false);
    d0 = __builtin_amdgcn_wmma_f32_16x16x4_f32(false, A1, false, B01, (short)0, d0, false, false);
    d1 = __builtin_amdgcn_wmma_f32_16x16x4_f32(false, A0, false, B10, (short)0, d1, false, false);
    d1 = __builtin_amdgcn_wmma_f32_16x16x4_f32(false, A1, false, B11, (short)0, d1, false, false);

    // D layout: d[i] = element (M=i, N=lane) on lanes 0-15 (tile column set).
    #pragma unroll
    for (int i = 0; i < NS; ++i) {
      float d1s = __shfl_xor(d1[i], 16, 32);
      float r = lo ? d0[i] : d1s;
      if (lane >= o) s[i] += r;
    }
  }

  // ---- Phase 3: replay chunk with true entry state, emit outputs. ----
  float z[NS];
  #pragma unroll
  for (int i = 0; i < NS; ++i) z[i] = s[i];
  const int uend = (start + CHL < TEXT) ? (start + CHL) : TEXT;
  for (int u = start; u < uend; ++u) {
    const float xv = load_s(u);
    const float y = df2t_step(z, br, ar, b0, xv);
    if (MODE == 0) {
      if (u >= PPAD && u < PPAD + T0)      yrow[u - PPAD] = y;
      else if (u < PPAD)                   prow[u] = y;
      else                                 prow[PPAD + (u - (PPAD + T0))] = y;
    } else {
      const int tt = TEXT - 1 - u;        // un-reverse; trim pads
      if (tt >= PPAD && tt < PPAD + T0)   yrow[tt - PPAD] = y;
    }
  }
}

// ---------------------------------------------------------------------------
// Launch: setup (swizzled transition powers), forward pass, backward pass.
// d_ws layout:
//   [0 .. 640)           : Aswz (5 levels x 4 regs x 32 lanes, 2.5 KB)
//   [640 .. 640+2048*54) : y pad buffer (443 KB)
// ---------------------------------------------------------------------------
extern "C" void kernel_launch(void* const* d_in, const int* in_sizes, int n_in,
                              void* d_out, int out_size, void* d_ws, size_t ws_size,
                              hipStream_t stream) {
  (void)in_sizes; (void)n_in; (void)out_size; (void)ws_size;
  const float* x  = (const float*)d_in[0];
  const float* b  = (const float*)d_in[1];
  const float* a  = (const float*)d_in[2];
  const float* zi = (const float*)d_in[3];
  float* out  = (float*)d_out;
  float* ws   = (float*)d_ws;
  float* Aswz = ws;
  float* ypad = ws + ASWZ_FLOATS;

  hipLaunchKernelGGL(butter_setup, dim3(1), dim3(64), 0, stream, a, Aswz);

  const int wavesPerBlock = 8;
  dim3 grid(NROWS / wavesPerBlock), block(32 * wavesPerBlock);
  hipLaunchKernelGGL((filt_pass<0>), grid, block, 0, stream,
                     x, b, a, zi, out, ypad, Aswz);
  hipLaunchKernelGGL((filt_pass<1>), grid, block, 0, stream,
                     x, b, a, zi, out, ypad, Aswz);
}